// Bilstm_CRF_53017076302088
// MI455X (gfx1250) — compile-verified
//
#include <hip/hip_runtime.h>

typedef __attribute__((ext_vector_type(2))) float v2f;
typedef __attribute__((ext_vector_type(8))) float v8f;

#define Bsz 512
#define Lsz 1024
#define Tsz 52           // TAGSET + 2
#define TP  64           // padded tag dim (multiple of 16)
#define START_TAG 50
#define STOP_TAG  51

// -------------------------------------------------------------------------
// CRF forward algorithm (log-partition) via exp-trick GEMM:
//   m_b     = max_i part[b,i]
//   p'[b,i] = exp(part[b,i] - m_b)
//   q[b,j]  = sum_i p'[b,i] * expT[i,j]         <-- V_WMMA_F32_16X16X4_F32
//   part    = mask ? log(q) + m_b + feats[t,b,j] : part
//
// 32 workgroups x 16 batch rows; 4 waves/WG, each wave owns a 16-wide N-tile
// and runs the K=64 reduction as 16 WMMA f32 16x16x4 ops per timestep.
// feats/mask for step t are loaded at the TOP of the iteration so the HBM
// latency of the scattered gathers overlaps phase A (max+exp) and the WMMA
// chain instead of serializing after them.
// -------------------------------------------------------------------------
__global__ __launch_bounds__(128) void crf_forward_kernel(
    const float* __restrict__ feats,          // (B, L, T)
    const float* __restrict__ trans,          // (T, T)
    const unsigned char* __restrict__ mask,   // (B, L) bool (1 byte)
    float* __restrict__ fwd)                  // (B,) per-row forward score
{
  __shared__ float expT_s[TP][80];   // stride 80 -> conflict-free B-row reads
  __shared__ float part_s[16][68];   // stride 68 -> conflict-free A-col reads
  __shared__ float pexp_s[16][68];
  __shared__ float mrow_s[16];
  __shared__ float red_s[128];

  const int tid    = threadIdx.x;
  const int wgbase = blockIdx.x * 16;          // first batch row of this WG
  const int lane   = tid & 31;
  const int wv     = tid >> 5;                 // wave id 0..3 = N-tile
  const int hi     = lane >> 4;                // lane half (K split)
  const int ln     = lane & 15;

  // exp(transitions) into LDS, zero-padded to 64x64
  for (int idx = tid; idx < TP * TP; idx += 128) {
    int i = idx >> 6, j = idx & 63;
    expT_s[i][j] = (i < Tsz && j < Tsz) ? __expf(trans[i * Tsz + j]) : 0.f;
  }
  // init: part[b,j] = feats[b,0,j] + trans[START,j]
  for (int idx = tid; idx < 16 * TP; idx += 128) {
    int r = idx >> 6, j = idx & 63;
    int b = wgbase + r;
    part_s[r][j] = (j < Tsz)
        ? feats[((size_t)b * Lsz) * Tsz + j] + trans[START_TAG * Tsz + j]
        : -3.4e38f;
  }
  __syncthreads();

  const int jcol = wv * 16 + ln;               // this lane's tag column
  const bool jvalid = (jcol < Tsz);

  for (int t = 1; t <= Lsz; ++t) {
    // ---- prefetch feats/mask for step t (latency hidden by A+B phases) ----
    float featv[8];
    unsigned char maskv[8];
    if (t < Lsz && jvalid) {
#pragma unroll
      for (int r = 0; r < 8; ++r) {
        int bg = wgbase + r + hi * 8;
        featv[r] = __builtin_nontemporal_load(
            &feats[((size_t)bg * Lsz + t) * Tsz + jcol]);
        maskv[r] = mask[(size_t)bg * Lsz + t];
      }
    }

    // ---- phase A: per-row max, then p' = exp(part - m) ----
    {
      int row = tid >> 3, sub = tid & 7;       // 8 threads per row
      float mx = -3.4e38f;
      for (int j = sub; j < Tsz; j += 8) mx = fmaxf(mx, part_s[row][j]);
      red_s[tid] = mx;
    }
    __syncthreads();
    if (tid < 16) {
      float m = red_s[tid * 8];
      for (int k = 1; k < 8; ++k) m = fmaxf(m, red_s[tid * 8 + k]);
      mrow_s[tid] = m;
    }
    __syncthreads();
    for (int idx = tid; idx < 16 * TP; idx += 128) {
      int r = idx >> 6, j = idx & 63;
      pexp_s[r][j] = (j < Tsz) ? __expf(part_s[r][j] - mrow_s[r]) : 0.f;
    }
    __syncthreads();

    if (t == Lsz) break;   // final p'/m feed the termination below

    // ---- phase B: q = p' @ expT  (16 x wmma_f32_16x16x4, K = 0..63) ----
    // A 16x4 f32: lane l -> M = l%16, K = 2*(l/16) + vgpr
    // B  4x16 f32: lane l -> N = l%16, K = 2*(l/16) + vgpr
    // (EXEC is all-1s here: no enclosing divergence.)
    v8f c = {0.f, 0.f, 0.f, 0.f, 0.f, 0.f, 0.f, 0.f};
#pragma unroll
    for (int kk = 0; kk < 16; ++kk) {
      int k0 = kk * 4 + hi * 2;
      v2f a, b;
      a.x = pexp_s[ln][k0];
      a.y = pexp_s[ln][k0 + 1];
      b.x = expT_s[k0][wv * 16 + ln];
      b.y = expT_s[k0 + 1][wv * 16 + ln];
      c = __builtin_amdgcn_wmma_f32_16x16x4_f32(
              false, a, false, b, (short)0, c, false, false);
    }

    // ---- phase C: masked update part = log(q) + m + feats[t] ----
    // C/D layout: vgpr r, lane l -> M = r + 8*(l/16), N = l%16
    if (jvalid) {
#pragma unroll
      for (int r = 0; r < 8; ++r) {
        int bl = r + hi * 8;
        if (maskv[r]) {
          part_s[bl][jcol] = __logf(c[r]) + mrow_s[bl] + featv[r];
        }
      }
    }
    __syncthreads();
  }

  // termination: fwd[b] = m_b + log(sum_i p'[b,i] * expT[i, STOP])
  if (tid < 16) {
    float s = 0.f;
    for (int i = 0; i < Tsz; ++i) s += pexp_s[tid][i] * expT_s[i][STOP_TAG];
    fwd[wgbase + tid] = mrow_s[tid] + __logf(s);
  }
}

// -------------------------------------------------------------------------
// Gold path score: one thread per batch row (prefix mask => simple chain).
// -------------------------------------------------------------------------
__global__ __launch_bounds__(128) void crf_gold_kernel(
    const float* __restrict__ feats,
    const float* __restrict__ trans,
    const unsigned char* __restrict__ mask,
    const int* __restrict__ tags,
    float* __restrict__ gold)
{
  int b = blockIdx.x * 128 + threadIdx.x;
  if (b >= Bsz) return;
  const unsigned char* mrow = mask + (size_t)b * Lsz;
  const int*           trow = tags + (size_t)b * Lsz;
  const float*         frow = feats + (size_t)b * Lsz * Tsz;
  int prev = START_TAG, last = 0;
  float s = 0.f;
  for (int t = 0; t < Lsz; ++t) {
    if (mrow[t]) {
      int tg = trow[t];
      s += frow[(size_t)t * Tsz + tg] + trans[prev * Tsz + tg];
      prev = tg;
      last = tg;
    }
  }
  s += trans[last * Tsz + STOP_TAG];
  gold[b] = s;
}

// -------------------------------------------------------------------------
// Deterministic final reduction: (sum fwd - sum gold) / B
// -------------------------------------------------------------------------
__global__ __launch_bounds__(512) void crf_reduce_kernel(
    const float* __restrict__ fwd, const float* __restrict__ gold,
    float* __restrict__ out)
{
  __shared__ float sd[Bsz];
  int tid = threadIdx.x;
  sd[tid] = fwd[tid] - gold[tid];
  __syncthreads();
  for (int st = Bsz / 2; st > 0; st >>= 1) {
    if (tid < st) sd[tid] += sd[tid + st];
    __syncthreads();
  }
  if (tid == 0) out[0] = sd[0] * (1.0f / (float)Bsz);
}

extern "C" void kernel_launch(void* const* d_in, const int* in_sizes, int n_in,
                              void* d_out, int out_size, void* d_ws, size_t ws_size,
                              hipStream_t stream) {
  const float*         feats = (const float*)d_in[0];          // (B,L,T) f32
  const float*         trans = (const float*)d_in[1];          // (T,T)   f32
  const unsigned char* mask  = (const unsigned char*)d_in[2];  // (B,L)   bool
  const int*           tags  = (const int*)d_in[3];            // (B,L)   i32
  float* out  = (float*)d_out;
  float* fwd  = (float*)d_ws;          // B floats
  float* gold = fwd + Bsz;             // B floats

  hipLaunchKernelGGL(crf_forward_kernel, dim3(Bsz / 16), dim3(128), 0, stream,
                     feats, trans, mask, fwd);
  hipLaunchKernelGGL(crf_gold_kernel, dim3((Bsz + 127) / 128), dim3(128), 0, stream,
                     feats, trans, mask, tags, gold);
  hipLaunchKernelGGL(crf_reduce_kernel, dim3(1), dim3(Bsz), 0, stream,
                     fwd, gold, out);
}